// ClientGCNConv_10703058501715
// MI455X (gfx1250) — compile-verified
//
#include <hip/hip_runtime.h>
#include <math.h>
#include <stddef.h>

// ---------------------------------------------------------------------------
// GCN scatter-max:  out[n, :] = max over edges e with dst[e]==n of
//                     f[src[e], :] * (norm[src[e]] * norm[dst[e]])
//                   nodes with no in-edges -> 0
//
// MI455X reasoning:
//  * Working set (f 12.8MB, out 12.8MB, norm 0.2MB, idx 6.4MB) is resident in
//    the 192MB L2 -> pure traffic bound is ~20us; the real limiter is the
//    51.2M per-dword f32 max RMWs at the L2 atomic units (~250us at ~200G/s).
//  * Fix: monotone test-and-atomic. Read the current value with a DEV-scope
//    load (bypasses the read-only WGP$, serviced by L2 where atomics commit)
//    and only issue global_atomic_max_num_f32 when the candidate exceeds it.
//    Always correct (stored values only grow; stale reads are <= current),
//    and cuts committed RMWs from ~16/elem toward ~H(16)=3.4/elem.
//  * WMMA is inapplicable: max is not on the WMMA semiring.
// ---------------------------------------------------------------------------

#define THREADS 256
#define LANES_PER_EDGE 16   // 16 lanes x float4 = 64 features per edge

// Native CDNA5 float-max atomic at device scope (RMW performed at L2).
__device__ __forceinline__ void atomic_max_f32(float* addr, float val) {
    asm volatile("global_atomic_max_num_f32 %0, %1, off scope:SCOPE_DEV"
                 :: "v"(addr), "v"(val)
                 : "memory");
}

// DEV-scope b128 load: cannot hit WGP-scope caches, so it observes the
// freshest L2 value (where the max atomics are performed). The embedded
// s_wait_loadcnt covers this load (the compiler doesn't track asm loads).
__device__ __forceinline__ float4 load_f32x4_dev(const float* addr) {
    float4 r;
    asm volatile("global_load_b128 %0, %1, off scope:SCOPE_DEV\n\t"
                 "s_wait_loadcnt 0x0"
                 : "=v"(r) : "v"(addr) : "memory");
    return r;
}

__global__ __launch_bounds__(THREADS)
void init_out_kernel(float4* __restrict__ out, int n4) {
    int i = blockIdx.x * THREADS + threadIdx.x;
    if (i < n4) {
        out[i] = make_float4(-INFINITY, -INFINITY, -INFINITY, -INFINITY);
    }
}

__global__ __launch_bounds__(THREADS)
void scatter_max_kernel(const float* __restrict__ f,
                        const float* __restrict__ norm,
                        const int*   __restrict__ src,
                        const int*   __restrict__ dst,
                        float*       __restrict__ out,
                        unsigned*    __restrict__ deg,
                        int E) {
    int t   = blockIdx.x * THREADS + threadIdx.x;
    int e   = t >> 4;           // edge index (16 lanes per edge)
    int sub = t & 15;           // which float4 chunk of the 64-float row
    if (e >= E) return;

    // Index streams are touched exactly once per launch: non-temporal so the
    // 2x3.2MB streams don't displace the L2-resident f/out arrays.
    int s = __builtin_nontemporal_load(src + e);
    int d = __builtin_nontemporal_load(dst + e);

    float w = norm[s] * norm[d];

    // One b128 gather per lane; 16 lanes cover the full 256B row, coalesced.
    // f is read-only -> normal (WGP$-cacheable) load is ideal.
    const float4 v = *reinterpret_cast<const float4*>(f + ((size_t)s << 6) + (sub << 2));

    float mx = v.x * w;
    float my = v.y * w;
    float mz = v.z * w;
    float mw = v.w * w;

    float* p = out + ((size_t)d << 6) + (sub << 2);

    // Monotone filter: only commit an RMW if we might raise the max.
    float4 cur = load_f32x4_dev(p);
    if (mx > cur.x) atomic_max_f32(p + 0, mx);
    if (my > cur.y) atomic_max_f32(p + 1, my);
    if (mz > cur.z) atomic_max_f32(p + 2, mz);
    if (mw > cur.w) atomic_max_f32(p + 3, mw);

    // In-degree count for exact has_in semantics (one atomic per edge).
    if (deg != nullptr && sub == 0) {
        atomicAdd(deg + d, 1u);
    }
}

// Exact reference semantics: rows with zero in-degree become 0.
__global__ __launch_bounds__(THREADS)
void finalize_deg_kernel(float4* __restrict__ out,
                         const unsigned* __restrict__ deg,
                         int N) {
    int t    = blockIdx.x * THREADS + threadIdx.x;
    int node = t >> 4;
    if (node >= N) return;
    if (deg[node] == 0u) {
        out[t] = make_float4(0.f, 0.f, 0.f, 0.f);
    }
}

// Fallback if workspace is too small for the degree array: rewrite the -inf
// sentinel to 0 (equivalent for any finite edge messages).
__global__ __launch_bounds__(THREADS)
void finalize_sentinel_kernel(float4* __restrict__ out, int n4) {
    int i = blockIdx.x * THREADS + threadIdx.x;
    if (i >= n4) return;
    float4 v = out[i];
    v.x = (v.x == -INFINITY) ? 0.f : v.x;
    v.y = (v.y == -INFINITY) ? 0.f : v.y;
    v.z = (v.z == -INFINITY) ? 0.f : v.z;
    v.w = (v.w == -INFINITY) ? 0.f : v.w;
    out[i] = v;
}

extern "C" void kernel_launch(void* const* d_in, const int* in_sizes, int n_in,
                              void* d_out, int out_size, void* d_ws, size_t ws_size,
                              hipStream_t stream) {
    // Inputs (setup_inputs order): f [N*64] f32, norm [N] f32,
    //                              src [E] int, dst [E] int, num_nodes [1] int
    const float* f    = (const float*)d_in[0];
    const float* norm = (const float*)d_in[1];
    const int*   src  = (const int*)d_in[2];
    const int*   dst  = (const int*)d_in[3];
    float*       out  = (float*)d_out;

    const int N = in_sizes[1];          // norm has N elements
    const int E = in_sizes[2];
    const int n4 = N * (64 / 4);        // float4 chunks in out

    const bool use_deg = ws_size >= (size_t)N * sizeof(unsigned);
    unsigned* deg = use_deg ? (unsigned*)d_ws : nullptr;

    // 1) init: out = -inf everywhere, deg = 0
    init_out_kernel<<<(n4 + THREADS - 1) / THREADS, THREADS, 0, stream>>>(
        (float4*)out, n4);
    if (use_deg) {
        hipMemsetAsync(deg, 0, (size_t)N * sizeof(unsigned), stream);
    }

    // 2) scatter-max over all edges, 16 lanes per edge
    long total  = (long)E * LANES_PER_EDGE;
    long blocks = (total + THREADS - 1) / THREADS;
    scatter_max_kernel<<<(int)blocks, THREADS, 0, stream>>>(
        f, norm, src, dst, out, deg, E);

    // 3) zero rows with no in-edges
    if (use_deg) {
        finalize_deg_kernel<<<(n4 + THREADS - 1) / THREADS, THREADS, 0, stream>>>(
            (float4*)out, deg, N);
    } else {
        finalize_sentinel_kernel<<<(n4 + THREADS - 1) / THREADS, THREADS, 0, stream>>>(
            (float4*)out, n4);
    }
}